// GCNConv_6846177869848
// MI455X (gfx1250) — compile-verified
//
#include <hip/hip_runtime.h>

typedef __attribute__((ext_vector_type(2))) float v2f;
typedef __attribute__((ext_vector_type(8))) float v8f;

#define IN_C 128
#define OUT_C 128

static inline unsigned cdiv_u(long long a, long long b) { return (unsigned)((a + b - 1) / b); }

// Hardware fire-and-forget f32 atomic add at device scope (L2 RMW, no return,
// tracked by STOREcnt; S_ENDPGM's implicit wait-idle drains it).
__device__ __forceinline__ void atomic_add_f32_hw(float* addr, float v) {
    asm volatile("global_atomic_add_f32 %0, %1, off scope:SCOPE_DEV"
                 :: "v"(addr), "v"(v)
                 : "memory");
}

// ---------------- degree kernels ----------------
__global__ void k_init_deg(int* __restrict__ deg, int n) {
    int i = blockIdx.x * blockDim.x + threadIdx.x;
    if (i < n) deg[i] = 1;  // self loop contributes 1
}

__global__ void k_count_deg(const int* __restrict__ rows, int* __restrict__ deg, int e) {
    int i = blockIdx.x * blockDim.x + threadIdx.x;
    if (i < e) atomicAdd(&deg[rows[i]], 1);
}

__global__ void k_rsqrt(const int* __restrict__ deg, float* __restrict__ dinv, int n) {
    int i = blockIdx.x * blockDim.x + threadIdx.x;
    if (i < n) dinv[i] = rsqrtf((float)deg[i]);
}

// ---------------- h = x @ W^T + b  via V_WMMA_F32_16X16X4_F32 ----------------
// One wave computes a 16x16 tile of h. Block = 256 threads = 8 waves = all 8
// column tiles (OUT_C=128) of one 16-row tile. K is walked in steps of 4.
//
// A (16x4 f32, ISA 7.12.2): lanes 0-15 hold M=lane, K={kk,kk+1}; lanes 16-31
// hold M=lane-16, K={kk+2,kk+3}  -> per-lane v2f at x[m*128 + kk + 2*half].
// B (4x16 f32): mirrored with N=lane%16, same K halves -> W[n*128 + kk + 2*half]
// (B[k][n] = W[n][k] since h = x @ W^T).
// C/D (16x16 f32): VGPR i holds M = i + 8*(lane>=16), N = lane%16.
__global__ __launch_bounds__(256) void k_gemm_wmma(const float* __restrict__ x,
                                                   const float* __restrict__ Wm,
                                                   const float* __restrict__ bias,
                                                   float* __restrict__ h, int n) {
    const int lane  = threadIdx.x & 31;
    const int wave  = threadIdx.x >> 5;
    const int half  = lane >> 4;
    const int l     = lane & 15;
    const int mBase = blockIdx.x * 16;
    const int nBase = wave * 16;

    int mrow = mBase + l;
    if (mrow > n - 1) mrow = n - 1;  // branchless tail clamp (EXEC stays all-ones)
    const int koff = half * 2;

    const float* xr = x  + (size_t)mrow * IN_C + koff;
    const float* wr = Wm + (size_t)(nBase + l) * IN_C + koff;

    v8f c = {0.f, 0.f, 0.f, 0.f, 0.f, 0.f, 0.f, 0.f};
#pragma unroll 8
    for (int kk = 0; kk < IN_C; kk += 4) {
        v2f a = *(const v2f*)(xr + kk);
        v2f b = *(const v2f*)(wr + kk);
        c = __builtin_amdgcn_wmma_f32_16x16x4_f32(false, a, false, b, (short)0, c,
                                                  false, false);
    }

    const float bv = bias[nBase + l];
    float* hp = h + (size_t)mBase * OUT_C + nBase + l;
#pragma unroll
    for (int i = 0; i < 8; ++i) {
        const int mi = i + half * 8;
        if (mBase + mi < n) hp[(size_t)mi * OUT_C] = c[i] + bv;
    }
}

// ---------------- self-loop term: out[i,:] = dinv[i]^2 * h[i,:] ----------------
__global__ void k_out_init(const float* __restrict__ h, const float* __restrict__ dinv,
                           float* __restrict__ out, int total) {
    int i = blockIdx.x * blockDim.x + threadIdx.x;
    if (i < total) {
        float d = dinv[i >> 7];
        out[i] = d * d * h[i];
    }
}

// ---------------- edge scatter: out[row,:] += dinv[row]*dinv[col]*h[col,:] -----
// One wave per edge: the edge id is forced into an SGPR via readfirstlane so
// rows/cols/dinv loads become scalar (1 per wave instead of 32 redundant lane
// loads). Each lane gathers 4 channels of h[col] with one b128 load (the wave
// reads one full 512B row, perfectly coalesced) and issues 4 hardware f32
// atomics into out[row] -- the minimum atomic count (no packed f32 atomic
// exists). h and out both live in the 192MB L2.
__global__ __launch_bounds__(256) void k_scatter(const int* __restrict__ rows,
                                                 const int* __restrict__ cols,
                                                 const float* __restrict__ dinv,
                                                 const float* __restrict__ h,
                                                 float* __restrict__ out, int e) {
    const int lane = threadIdx.x & 31;
    const int wave = threadIdx.x >> 5;
    const int ed   = __builtin_amdgcn_readfirstlane(blockIdx.x * 8 + wave);
    if (ed >= e) return;                      // wave-uniform exit

    const int   r   = rows[ed];               // scalar loads (SGPR address)
    const int   cl  = cols[ed];
    const float nrm = dinv[r] * dinv[cl];

    const int c4 = lane << 2;                 // 4 channels per lane
    const float4 hv = *(const float4*)(h + (size_t)cl * OUT_C + c4);
    float* op = out + (size_t)r * OUT_C + c4;
    atomic_add_f32_hw(op + 0, nrm * hv.x);
    atomic_add_f32_hw(op + 1, nrm * hv.y);
    atomic_add_f32_hw(op + 2, nrm * hv.z);
    atomic_add_f32_hw(op + 3, nrm * hv.w);
}

extern "C" void kernel_launch(void* const* d_in, const int* in_sizes, int n_in,
                              void* d_out, int out_size, void* d_ws, size_t ws_size,
                              hipStream_t stream) {
    const float* x    = (const float*)d_in[0];
    const int*   ei   = (const int*)d_in[1];   // [2, E] flat, int32 per harness
    const float* Wm   = (const float*)d_in[2];
    const float* bias = (const float*)d_in[3];
    float*       out  = (float*)d_out;

    const int Nn = in_sizes[0] / IN_C;
    const int Ec = in_sizes[1] / 2;
    const int* rows = ei;
    const int* cols = ei + Ec;

    // workspace layout: h[N*128 f32] | deg[N i32] | dinv[N f32]
    char*  base = (char*)d_ws;
    float* h    = (float*)base;
    int*   deg  = (int*)(base + (size_t)Nn * OUT_C * sizeof(float));
    float* dinv = (float*)((char*)deg + (size_t)Nn * sizeof(int));

    k_init_deg<<<cdiv_u(Nn, 256), 256, 0, stream>>>(deg, Nn);
    k_count_deg<<<cdiv_u(Ec, 256), 256, 0, stream>>>(rows, deg, Ec);
    k_gemm_wmma<<<cdiv_u(Nn, 16), 256, 0, stream>>>(x, Wm, bias, h, Nn);
    k_rsqrt<<<cdiv_u(Nn, 256), 256, 0, stream>>>(deg, dinv, Nn);
    k_out_init<<<cdiv_u((long long)Nn * OUT_C, 256), 256, 0, stream>>>(h, dinv, out,
                                                                       Nn * OUT_C);
    k_scatter<<<cdiv_u(Ec, 8), 256, 0, stream>>>(rows, cols, dinv, h, out, Ec);
}